// AccumulativeGainLoss_61864708931686
// MI455X (gfx1250) — compile-verified
//
#include <hip/hip_runtime.h>
#include <hip/hip_bf16.h>
#include <math.h>

// Problem constants (from reference setup_inputs)
#define BB 16
#define TT 32
#define NN 6000
#define KK 32
#define DD 8
#define TU 4                     // timesteps batched per wave in k_fty
#define EPSF 1e-8f
#define VALUE_DECAY 0.9f
#define PENALTY_WEIGHT 0.1f

typedef float v2f __attribute__((ext_vector_type(2)));
typedef float v8f __attribute__((ext_vector_type(8)));

// Workspace layout (floats)
#define OFF_FTF   0                       // B*K*K        = 16384
#define OFF_SUMF  (OFF_FTF + BB*KK*KK)    // B*K          = 512
#define OFF_FTY   (OFF_SUMF + BB*KK)      // B*T*K*D      = 131072
#define OFF_SY    (OFF_FTY + BB*TT*KK*DD) // B*T*D        = 4096
#define OFF_SY2   (OFF_SY + BB*TT*DD)     // B*T*D        = 4096
#define OFF_PART  (OFF_SY2 + BB*TT*DD)    // B            = 16

static __device__ __forceinline__ v8f wmma_f32_4(v2f a, v2f b, v8f c) {
    // D = A(16x4,f32) * B(4x16,f32) + C(16x16,f32)
    return __builtin_amdgcn_wmma_f32_16x16x4_f32(
        /*neg_a=*/false, a, /*neg_b=*/false, b,
        /*c_mod=*/(short)0, c, /*reuse_a=*/false, /*reuse_b=*/false);
}

// ---------------------------------------------------------------------------
// Kernel A: per-batch FtF = preds^T preds  (32x32) and sumF (column sums).
// One wave per batch. WMMA f32 16x16x4, 4 tiles, reducing N in chunks of 4.
// Lane (m + 16h), vgpr v holds preds[n0 + 2h + v, k0 + m]; this register set
// is simultaneously the A operand (A[m, 2h+v]) and the B operand (B[2h+v, m]).
// ---------------------------------------------------------------------------
__global__ __launch_bounds__(32)
void k_ftf(const float* __restrict__ preds, float* __restrict__ ftf,
           float* __restrict__ sumF) {
    const int b    = blockIdx.x;
    const int lane = threadIdx.x;
    const int h    = lane >> 4;   // half-wave id
    const int m    = lane & 15;
    const float* P = preds + (size_t)b * NN * KK;

    v8f c00 = {}, c01 = {}, c10 = {}, c11 = {};
    float s0 = 0.f, s1 = 0.f;

    for (int n0 = 0; n0 < NN; n0 += 4) {
        const float* r = P + (size_t)(n0 + 2 * h) * KK;
        v2f t0, t1;
        t0.x = r[m];          t0.y = r[KK + m];        // k-tile 0, rows n0+2h, n0+2h+1
        t1.x = r[16 + m];     t1.y = r[KK + 16 + m];   // k-tile 1
        s0 += t0.x + t0.y;
        s1 += t1.x + t1.y;
        c00 = wmma_f32_4(t0, t0, c00);
        c01 = wmma_f32_4(t0, t1, c01);
        c10 = wmma_f32_4(t1, t0, c10);
        c11 = wmma_f32_4(t1, t1, c11);
    }

    // column sums: combine the two half-waves (rows n≡{0,1} vs {2,3} mod 4)
    float o0 = s0 + __shfl_xor(s0, 16, 32);
    float o1 = s1 + __shfl_xor(s1, 16, 32);
    if (h == 0) {
        sumF[b * KK + m]      = o0;
        sumF[b * KK + 16 + m] = o1;
    }

    // C/D layout: vgpr v, low lanes -> M=v, high lanes -> M=v+8; N = lane&15
    float* F = ftf + (size_t)b * KK * KK;
#pragma unroll
    for (int v = 0; v < 8; ++v) {
        const int M = v + 8 * h;
        F[M * KK + m]             = c00[v];
        F[M * KK + 16 + m]        = c01[v];
        F[(16 + M) * KK + m]      = c10[v];
        F[(16 + M) * KK + 16 + m] = c11[v];
    }
}

// ---------------------------------------------------------------------------
// Kernel B: Fty[b,t,k,d] = sum_n preds[b,n,k]*y[b,t,n,d] (32x8 per (b,t)),
// fused with sum_n y and sum_n y^2. One wave handles TU=4 timesteps so the
// preds (A) fragments are loaded once and reused for 4 B fragments:
// 12 loads : 8 WMMA per 4-row chunk, and 4x less L2 traffic on preds.
// D=8 padded to the 16-wide WMMA B operand with zeros in lanes j>=8.
// ---------------------------------------------------------------------------
__global__ __launch_bounds__(32)
void k_fty(const float* __restrict__ preds, const float* __restrict__ y,
           float* __restrict__ fty, float* __restrict__ sy,
           float* __restrict__ sy2) {
    const int blk  = blockIdx.x;            // B * T/TU blocks
    const int b    = blk / (TT / TU);
    const int t0   = (blk % (TT / TU)) * TU;
    const int lane = threadIdx.x;
    const int h    = lane >> 4;
    const int j    = lane & 15;
    const int jd   = j & 7;                 // clamped (OOB-safe) d index
    const float dm = (j < 8) ? 1.0f : 0.0f;

    const float* P = preds + (size_t)b * NN * KK;
    const float* Y0 = y + ((size_t)(b * TT + t0)) * NN * DD;

    v8f c0[TU], c1[TU];
    float a_sy[TU], a_sy2[TU];
#pragma unroll
    for (int u = 0; u < TU; ++u) {
        c0[u] = (v8f){};  c1[u] = (v8f){};
        a_sy[u] = 0.f;    a_sy2[u] = 0.f;
    }

    for (int n0 = 0; n0 < NN; n0 += 4) {
        const float* pr = P + (size_t)(n0 + 2 * h) * KK;
        v2f a0, a1;
        a0.x = pr[j];          a0.y = pr[KK + j];
        a1.x = pr[16 + j];     a1.y = pr[KK + 16 + j];
        const size_t yro = (size_t)(n0 + 2 * h) * DD;
#pragma unroll
        for (int u = 0; u < TU; ++u) {
            const float* yr = Y0 + (size_t)u * NN * DD + yro;
            v2f bv;
            bv.x = yr[jd] * dm;          // B[2h+0, j] = y[t0+u, n0+2h,   j]
            bv.y = yr[DD + jd] * dm;     // B[2h+1, j] = y[t0+u, n0+2h+1, j]
            a_sy[u]  += bv.x + bv.y;
            a_sy2[u] += bv.x * bv.x + bv.y * bv.y;
            c0[u] = wmma_f32_4(a0, bv, c0[u]);   // k-tile 0 rows of Fty
            c1[u] = wmma_f32_4(a1, bv, c1[u]);   // k-tile 1
        }
    }

#pragma unroll
    for (int u = 0; u < TU; ++u) {
        const int bt = b * TT + t0 + u;
        float s1 = a_sy[u]  + __shfl_xor(a_sy[u], 16, 32);
        float s2 = a_sy2[u] + __shfl_xor(a_sy2[u], 16, 32);
        if (h == 0 && j < 8) {
            sy[bt * DD + j]  = s1;
            sy2[bt * DD + j] = s2;
        }
        if (j < 8) {
            float* Fo = fty + (size_t)bt * KK * DD;
#pragma unroll
            for (int v = 0; v < 8; ++v) {
                const int M = v + 8 * h;
                Fo[M * DD + j]        = c0[u][v];
                Fo[(16 + M) * DD + j] = c1[u][v];
            }
        }
    }
}

// ---------------------------------------------------------------------------
// Kernel C: per-batch small linear algebra. One wave per batch.
//  - Gauss-Jordan inverse of (FtF + eps I) in LDS
//  - for each t: beta = inv * Fty ; ss_res/ss_tot via small quadratic forms
//  - correlation penalty from FtF and sumF
// ---------------------------------------------------------------------------
__global__ __launch_bounds__(32)
void k_stage3(const float* __restrict__ ftf, const float* __restrict__ sumF,
              const float* __restrict__ fty, const float* __restrict__ sy,
              const float* __restrict__ sy2,
              const float* __restrict__ importance,
              float* __restrict__ partial) {
    const int b    = blockIdx.x;
    const int lane = threadIdx.x;     // 0..31, one row per lane

    __shared__ float Ai[KK][KK];      // working matrix / later raw FtF copy
    __shared__ float Inv[KK][KK];
    __shared__ float FY[KK][DD];
    __shared__ float BE[KK][DD];
    __shared__ float STD[KK];
    __shared__ float SF[KK];

    const float* G = ftf + (size_t)b * KK * KK;
    for (int c = 0; c < KK; ++c) {
        Ai[lane][c]  = G[lane * KK + c] + ((c == lane) ? EPSF : 0.0f);
        Inv[lane][c] = (c == lane) ? 1.0f : 0.0f;
    }
    SF[lane] = sumF[b * KK + lane];
    __syncthreads();

    // Gauss-Jordan (SPD + ridge => no pivoting)
    for (int k = 0; k < KK; ++k) {
        const float ip = 1.0f / Ai[k][k];
        __syncthreads();
        Ai[k][lane]  *= ip;          // each lane scales one column of row k
        Inv[k][lane] *= ip;
        __syncthreads();
        if (lane != k) {
            const float f = Ai[lane][k];
            for (int c = 0; c < KK; ++c) {
                Ai[lane][c]  -= f * Ai[k][c];
                Inv[lane][c] -= f * Inv[k][c];
            }
        }
        __syncthreads();
    }

    // restore raw FtF (no eps) for the quadratic form + correlation
    for (int c = 0; c < KK; ++c) Ai[lane][c] = G[lane * KK + c];
    __syncthreads();

    float imp[DD];
#pragma unroll
    for (int d = 0; d < DD; ++d) imp[d] = importance[d];

    const float invN = 1.0f / (float)NN;
    float total_r2 = 0.0f;
    float decay = 1.0f;

    for (int t = 0; t < TT; ++t) {
        const float* Ft = fty + ((size_t)(b * TT + t)) * KK * DD;
#pragma unroll
        for (int d = 0; d < DD; ++d) FY[lane][d] = Ft[lane * DD + d];
        __syncthreads();

        // beta[lane][d] = sum_j Inv[lane][j] * FY[j][d]
        float beta[DD];
#pragma unroll
        for (int d = 0; d < DD; ++d) beta[d] = 0.0f;
        for (int jj = 0; jj < KK; ++jj) {
            const float w = Inv[lane][jj];
#pragma unroll
            for (int d = 0; d < DD; ++d) beta[d] += w * FY[jj][d];
        }
#pragma unroll
        for (int d = 0; d < DD; ++d) BE[lane][d] = beta[d];
        __syncthreads();

        // p1[d] = sum_k beta[k][d]*Fty[k][d]; p2[d] = beta^T FtF beta
        float p1[DD], p2[DD];
#pragma unroll
        for (int d = 0; d < DD; ++d) { p1[d] = beta[d] * FY[lane][d]; p2[d] = 0.0f; }
        for (int jj = 0; jj < KK; ++jj) {
            const float g = Ai[lane][jj];
#pragma unroll
            for (int d = 0; d < DD; ++d) p2[d] += g * BE[jj][d];
        }
#pragma unroll
        for (int d = 0; d < DD; ++d) p2[d] *= beta[d];

        for (int off = 16; off >= 1; off >>= 1) {
#pragma unroll
            for (int d = 0; d < DD; ++d) {
                p1[d] += __shfl_xor(p1[d], off, 32);
                p2[d] += __shfl_xor(p2[d], off, 32);
            }
        }

        float wr2 = 0.0f;
#pragma unroll
        for (int d = 0; d < DD; ++d) {
            const float syv  = sy[(b * TT + t) * DD + d];
            const float sy2v = sy2[(b * TT + t) * DD + d];
            const float ss_res = sy2v - 2.0f * p1[d] + p2[d];
            const float ss_tot = sy2v - syv * syv * invN + EPSF;
            wr2 += imp[d] * (1.0f - ss_res / ss_tot);
        }
        total_r2 += wr2 * decay;
        decay *= VALUE_DECAY;
        __syncthreads();   // done reading FY/BE before next t overwrites
    }

    // correlation penalty: cov = FtF - sumF sumF^T / N
    STD[lane] = sqrtf(Ai[lane][lane] - SF[lane] * SF[lane] * invN);
    __syncthreads();
    float cp = 0.0f;
    const float sfk = SF[lane], stdk = STD[lane];
    for (int jj = 0; jj < KK; ++jj) {
        if (jj != lane) {
            const float cov  = Ai[lane][jj] - sfk * SF[jj] * invN;
            const float corr = cov / (stdk * STD[jj]);
            cp += corr * corr;
        }
    }
    for (int off = 16; off >= 1; off >>= 1) cp += __shfl_xor(cp, off, 32);

    if (lane == 0)
        partial[b] = (-total_r2 / (float)TT + PENALTY_WEIGHT * cp) / (float)BB;
}

// ---------------------------------------------------------------------------
// Kernel D: final 16 -> 1 reduction.
// ---------------------------------------------------------------------------
__global__ __launch_bounds__(32)
void k_reduce(const float* __restrict__ partial, float* __restrict__ out) {
    const int lane = threadIdx.x;
    float v = (lane < BB) ? partial[lane] : 0.0f;
    for (int off = 16; off >= 1; off >>= 1) v += __shfl_xor(v, off, 32);
    if (lane == 0) out[0] = v;
}

extern "C" void kernel_launch(void* const* d_in, const int* in_sizes, int n_in,
                              void* d_out, int out_size, void* d_ws, size_t ws_size,
                              hipStream_t stream) {
    (void)in_sizes; (void)n_in; (void)out_size; (void)ws_size;
    const float* preds      = (const float*)d_in[0];   // [B,N,K] f32
    const float* y_ts       = (const float*)d_in[1];   // [B,T,N,D] f32
    const float* importance = (const float*)d_in[2];   // [D] f32
    float* out = (float*)d_out;
    float* ws  = (float*)d_ws;                          // needs ~625 KB

    float* ftf  = ws + OFF_FTF;
    float* sumF = ws + OFF_SUMF;
    float* fty  = ws + OFF_FTY;
    float* sy   = ws + OFF_SY;
    float* sy2  = ws + OFF_SY2;
    float* part = ws + OFF_PART;

    k_ftf   <<<BB,               32, 0, stream>>>(preds, ftf, sumF);
    k_fty   <<<BB * (TT / TU),   32, 0, stream>>>(preds, y_ts, fty, sy, sy2);
    k_stage3<<<BB,               32, 0, stream>>>(ftf, sumF, fty, sy, sy2, importance, part);
    k_reduce<<<1,                32, 0, stream>>>(part, out);
}